// GCNLayer_13529146982751
// MI455X (gfx1250) — compile-verified
//
#include <hip/hip_runtime.h>

typedef float v2f __attribute__((ext_vector_type(2)));
typedef float v8f __attribute__((ext_vector_type(8)));

// ---------------------------------------------------------------------------
// cnt[i] = 0, fill[i] = 0
__global__ __launch_bounds__(256) void k_zero(int* __restrict__ cnt,
                                              int* __restrict__ fill, int N) {
    int i = blockIdx.x * 256 + threadIdx.x;
    if (i < N) { cnt[i] = 0; fill[i] = 0; }
}

// cnt[dst[e]] += 1   (in-degree, excluding self loop)
__global__ __launch_bounds__(256) void k_count(const long long* __restrict__ dst,
                                               int* __restrict__ cnt, int E) {
    int e = blockIdx.x * 256 + threadIdx.x;
    if (e < E) atomicAdd(&cnt[(int)dst[e]], 1);
}

// dis[i] = rsqrt(cnt[i] + 1)   (+1 = self loop, so always > 0)
__global__ __launch_bounds__(256) void k_rsqrt(const int* __restrict__ cnt,
                                               float* __restrict__ dis, int N) {
    int i = blockIdx.x * 256 + threadIdx.x;
    if (i < N) dis[i] = rsqrtf((float)cnt[i] + 1.0f);
}

// Exclusive prefix sum: rowptr[0]=0, rowptr[i+1]=sum(cnt[0..i]).
// Single 256-thread block, chunked Hillis-Steele in LDS; N=10000 -> 40 chunks.
__global__ __launch_bounds__(256) void k_scan(const int* __restrict__ cnt,
                                              int* __restrict__ rowptr, int N) {
    __shared__ int s[256];
    int carry = 0;
    if (threadIdx.x == 0) rowptr[0] = 0;
    for (int base = 0; base < N; base += 256) {
        int i = base + threadIdx.x;
        int v = (i < N) ? cnt[i] : 0;
        s[threadIdx.x] = v;
        __syncthreads();
        #pragma unroll
        for (int ofs = 1; ofs < 256; ofs <<= 1) {
            int t = (threadIdx.x >= ofs) ? s[threadIdx.x - ofs] : 0;
            __syncthreads();
            s[threadIdx.x] += t;
            __syncthreads();
        }
        if (i < N) rowptr[i + 1] = carry + s[threadIdx.x];
        int total = s[255];
        __syncthreads();
        carry += total;
    }
}

// csr[rowptr[dst] + slot] = src   (slot via per-node int atomic)
__global__ __launch_bounds__(256) void k_fill(const long long* __restrict__ src,
                                              const long long* __restrict__ dst,
                                              const int* __restrict__ rowptr,
                                              int* __restrict__ fill,
                                              int* __restrict__ csr, int E) {
    int e = blockIdx.x * 256 + threadIdx.x;
    if (e >= E) return;
    int d = (int)dst[e];
    int slot = rowptr[d] + atomicAdd(&fill[d], 1);
    csr[slot] = (int)src[e];
}

// ---------------------------------------------------------------------------
// H = X @ W, D = 512 fixed. FP32 via V_WMMA_F32_16X16X4_F32.
// Block = 256 threads = 8 waves; block computes a full 16(M) x 512(N) row-tile.
// Wave w owns columns [w*64, w*64+64) with 4 independent 16x16 accumulators.
__global__ __launch_bounds__(256) void k_gemm_wmma(const float* __restrict__ X,
                                                   const float* __restrict__ W,
                                                   float* __restrict__ H,
                                                   int M) {
    constexpr int D = 512;
    __shared__ float As[16 * D];   // 32 KB, packed As[(k>>1)*32 + m*2 + (k&1)]
    const int tid  = threadIdx.x;
    const int lane = tid & 31;
    const int wave = tid >> 5;
    const int row0 = blockIdx.x * 16;
    const bool full = (row0 + 16 <= M);   // block-uniform

    #pragma unroll
    for (int i = 0; i < 8; ++i) {
        int idx = tid + i * 256;        // 0..2047
        int r   = idx >> 7;             // row 0..15
        int c4  = idx & 127;            // float4 col
        int rr  = row0 + r;
        if (!full && rr >= M) rr = M - 1;
        const float4 v = ((const float4*)X)[(unsigned)rr * (D / 4) + c4];
        int k  = c4 * 4;
        int b0 = ((k >> 1) << 5) + (r << 1);
        As[b0]      = v.x;  As[b0 + 1]  = v.y;
        As[b0 + 32] = v.z;  As[b0 + 33] = v.w;
    }
    __syncthreads();

    const int m    = lane & 15;
    const int khi  = (lane >> 4) << 1;             // 0 or 2 (ISA A layout)
    const int col0 = wave * 64 + m;

    const float* Wp = W  + (unsigned)khi * D + col0;
    const float* Ap = As + ((khi >> 1) << 5) + (m << 1);

    v8f c0 = {0.f,0.f,0.f,0.f,0.f,0.f,0.f,0.f};
    v8f c1 = c0, c2 = c0, c3 = c0;

    #pragma unroll 4
    for (int kb = 0; kb < D; kb += 4) {
        const v2f a = *(const v2f*)(Ap + kb * 16);
        const float* wk = Wp + kb * D;
        v2f b0v, b1v, b2v, b3v;
        b0v.x = wk[0];   b0v.y = wk[D];
        b1v.x = wk[16];  b1v.y = wk[D + 16];
        b2v.x = wk[32];  b2v.y = wk[D + 32];
        b3v.x = wk[48];  b3v.y = wk[D + 48];
        c0 = __builtin_amdgcn_wmma_f32_16x16x4_f32(false, a, false, b0v, (short)0, c0, false, false);
        c1 = __builtin_amdgcn_wmma_f32_16x16x4_f32(false, a, false, b1v, (short)0, c1, false, false);
        c2 = __builtin_amdgcn_wmma_f32_16x16x4_f32(false, a, false, b2v, (short)0, c2, false, false);
        c3 = __builtin_amdgcn_wmma_f32_16x16x4_f32(false, a, false, b3v, (short)0, c3, false, false);
    }

    const int hi8 = (lane >> 4) << 3;
    float* Hp = H + (unsigned)(row0 + hi8) * D + col0;
    if (full) {
        #pragma unroll
        for (int r = 0; r < 8; ++r) {
            Hp[r * D +  0] = c0[r];
            Hp[r * D + 16] = c1[r];
            Hp[r * D + 32] = c2[r];
            Hp[r * D + 48] = c3[r];
        }
    } else {
        #pragma unroll
        for (int r = 0; r < 8; ++r) {
            if (row0 + hi8 + r < M) {
                Hp[r * D +  0] = c0[r];
                Hp[r * D + 16] = c1[r];
                Hp[r * D + 32] = c2[r];
                Hp[r * D + 48] = c3[r];
            }
        }
    }
}

// ---------------------------------------------------------------------------
// CSR gather: out[i] = b + dis[i]*( dis[i]*h[i] + sum_in dis[s]*h[s] ).
// One 128-thread block (4 waves) per node; thread t owns float4 column t.
// All h rows and csr/dis are L2-resident; zero float atomics; out written once.
__global__ __launch_bounds__(128) void k_gather(const float* __restrict__ h,
                                                const float* __restrict__ bias,
                                                const float* __restrict__ dis,
                                                const int* __restrict__ rowptr,
                                                const int* __restrict__ csr,
                                                float* __restrict__ out) {
    const int i = blockIdx.x;
    const int t = threadIdx.x;                 // 0..127
    const float4* h4 = (const float4*)h;
    const float di = dis[i];

    float4 acc;                                 // dis[i]*h[i] (self loop term)
    {
        float4 hv = h4[(unsigned)i * 128u + t];
        acc.x = hv.x * di; acc.y = hv.y * di;
        acc.z = hv.z * di; acc.w = hv.w * di;
    }

    const int beg = rowptr[i];
    const int end = rowptr[i + 1];
    int j = beg;
    for (; j + 1 < end; j += 2) {              // 2-way pipelined gather
        int s0 = csr[j], s1 = csr[j + 1];
        float w0 = dis[s0], w1 = dis[s1];
        float4 v0 = h4[(unsigned)s0 * 128u + t];
        float4 v1 = h4[(unsigned)s1 * 128u + t];
        acc.x += v0.x * w0 + v1.x * w1;
        acc.y += v0.y * w0 + v1.y * w1;
        acc.z += v0.z * w0 + v1.z * w1;
        acc.w += v0.w * w0 + v1.w * w1;
    }
    if (j < end) {
        int s0 = csr[j];
        float w0 = dis[s0];
        float4 v0 = h4[(unsigned)s0 * 128u + t];
        acc.x += v0.x * w0; acc.y += v0.y * w0;
        acc.z += v0.z * w0; acc.w += v0.w * w0;
    }

    float4 bv = ((const float4*)bias)[t];
    float4 o;
    o.x = bv.x + di * acc.x;  o.y = bv.y + di * acc.y;
    o.z = bv.z + di * acc.z;  o.w = bv.w + di * acc.w;
    ((float4*)out)[(unsigned)i * 128u + t] = o;
}

// ---------------------------------------------------------------------------
extern "C" void kernel_launch(void* const* d_in, const int* in_sizes, int n_in,
                              void* d_out, int out_size, void* d_ws, size_t ws_size,
                              hipStream_t stream) {
    const float*     x  = (const float*)d_in[0];
    const long long* ei = (const long long*)d_in[1];   // int64 per reference
    const float*     W  = (const float*)d_in[2];
    const float*     b  = (const float*)d_in[3];
    float* out = (float*)d_out;

    const int D = in_sizes[3];          // 512 (kernels specialized for this)
    const int N = in_sizes[0] / D;      // 10000
    const int E = in_sizes[1] / 2;      // 160000

    // workspace layout (4-byte words): cnt[N] | fill[N] | rowptr[N+4] | csr[E]
    //                                  | dis[N] | h[N*D] (16B aligned)
    int*   cnt    = (int*)d_ws;
    int*   fill   = cnt + N;
    int*   rowptr = fill + N;
    int*   csr    = rowptr + (N + 4);
    float* dis    = (float*)(csr + E);
    size_t hoff   = (size_t)(4 * N + 4 + E);
    hoff = (hoff + 3) & ~(size_t)3;                 // float4 alignment for h
    float* h      = (float*)d_ws + hoff;

    const long long* e_src = ei;        // edge_index[0]
    const long long* e_dst = ei + E;    // edge_index[1]

    k_zero  <<<(N + 255) / 256, 256, 0, stream>>>(cnt, fill, N);
    k_count <<<(E + 255) / 256, 256, 0, stream>>>(e_dst, cnt, E);
    k_rsqrt <<<(N + 255) / 256, 256, 0, stream>>>(cnt, dis, N);
    k_scan  <<<1, 256, 0, stream>>>(cnt, rowptr, N);
    k_fill  <<<(E + 255) / 256, 256, 0, stream>>>(e_src, e_dst, rowptr, fill, csr, E);

    k_gemm_wmma<<<(N + 15) / 16, 256, 0, stream>>>(x, W, h, N);

    k_gather<<<N, 128, 0, stream>>>(h, b, dis, rowptr, csr, out);
}